// RCNN_40819369181728
// MI455X (gfx1250) — compile-verified
//
#include <hip/hip_runtime.h>
#include <math.h>

#define A_N    1000
#define LQ     2048
#define C_N    3
#define H_N    10
#define HID    10000   // A_N * H_N
#define MID    5000
#define NPOS   1001
#define NOUT   2001
#define KCAT   6001    // MID + NPOS
#define NSPLIT 32      // K-chunks per 16-row tile (memory-level parallelism)

typedef float v2f __attribute__((ext_vector_type(2)));
typedef float v8f __attribute__((ext_vector_type(8)));

__device__ __forceinline__ float sigmoidf_(float x) {
    return 1.0f / (1.0f + expf(-x));
}

// One "LSTM" step as used by the reference (hidden state is dead: gates = x @ w_ih.T + b_ih + b_hh).
// i = rows [0,10), g = rows [20,30), o = rows [30,40) of the (40,K) w_ih. f is unused.
__device__ __forceinline__ void lstm_layer(const float* __restrict__ w,
                                           const float* __restrict__ bi,
                                           const float* __restrict__ bh,
                                           const float* xin, int K,
                                           float* hout) {
    for (int j = 0; j < H_N; ++j) {
        float gi = bi[j]         + bh[j];
        float gg = bi[2*H_N + j] + bh[2*H_N + j];
        float go = bi[3*H_N + j] + bh[3*H_N + j];
        for (int c = 0; c < K; ++c) {
            float xv = xin[c];
            gi += w[j*K + c]           * xv;
            gg += w[(2*H_N + j)*K + c] * xv;
            go += w[(3*H_N + j)*K + c] * xv;
        }
        float cc = sigmoidf_(gi) * tanhf(gg);
        hout[j] = sigmoidf_(go) * tanhf(cc);
    }
}

// Stage 1: per-agent LSTM head -> conc[10000]; also copy position + zero pad into combined vec tail.
__global__ void lstm_head_kernel(const float* __restrict__ quotes,
                                 const float* __restrict__ position,
                                 const float* __restrict__ w0, const float* __restrict__ bi0, const float* __restrict__ bh0,
                                 const float* __restrict__ w1, const float* __restrict__ bi1, const float* __restrict__ bh1,
                                 const float* __restrict__ w2, const float* __restrict__ bi2, const float* __restrict__ bh2,
                                 float* __restrict__ conc,
                                 float* __restrict__ vtail) {
    int tid = blockIdx.x * blockDim.x + threadIdx.x;

    // combined vector tail: position followed by 3 zeros of padding
    if (tid < NPOS + 3) vtail[tid] = (tid < NPOS) ? position[tid] : 0.0f;

    if (tid >= A_N) return;

    float x[C_N];
    size_t base = (size_t)tid * LQ * C_N + (size_t)(LQ - 1) * C_N;
    x[0] = quotes[base + 0];
    x[1] = quotes[base + 1];
    x[2] = quotes[base + 2];

    float ha[H_N], hb[H_N];
    lstm_layer(w0, bi0, bh0, x,  C_N, ha);
    lstm_layer(w1, bi1, bh1, ha, H_N, hb);
    lstm_layer(w2, bi2, bh2, hb, H_N, ha);

    for (int j = 0; j < H_N; ++j)
        conc[tid * H_N + j] = tanhf(ha[j]);
}

// ---- WMMA helpers ----------------------------------------------------------

__device__ __forceinline__ v8f wmma_f32_16x16x4(v2f a, v2f b, v8f c) {
    return __builtin_amdgcn_wmma_f32_16x16x4_f32(
        /*neg_a=*/false, a, /*neg_b=*/false, b,
        /*c_mod=*/(short)0, c, /*reuse_a=*/false, /*reuse_b=*/false);
}

template <bool EVENK>
__device__ __forceinline__ v2f load_a_tile(const float* __restrict__ wrow, int kk) {
    v2f a;
    if (EVENK) {
        // Kt even -> row base even -> kk even -> 8B aligned: single b64 NT load
        a = __builtin_nontemporal_load((const v2f*)(wrow + kk));
    } else {
        a.x = __builtin_nontemporal_load(wrow + kk);
        a.y = __builtin_nontemporal_load(wrow + kk + 1);
    }
    return a;
}

// WMMA-based partial GEMV: one wave32 computes, for 16 output rows, the partial dot
// over one K-chunk using V_WMMA_F32_16X16X4_F32, and writes it to part[chunk][row].
//   A tile = 16x4 weight tile (lane m<16: row m, K=0,1; lane m+16: row m, K=2,3)
//   B tile = vec chunk replicated across all 16 columns -> every D column = GEMV result
//   Lanes 0 (M=0..7) and 16 (M=8..15) hold column 0 and store.
// Weights streamed non-temporally (single-use; keep vec resident in L2/WGP$).
// Manual unroll x4 keeps 4 independent b64 weight loads in flight per wave.
template <bool EVENK>
__global__ void gemv_wmma_partial(const float* __restrict__ W,
                                  const float* __restrict__ vec,
                                  float* __restrict__ part,
                                  int nrows, int Kt, int KC) {
    const int lane = threadIdx.x & 31;
    const int half = lane >> 4;          // 0: K offset 0, 1: K offset +2
    const int m    = lane & 15;          // row within the 16-row tile
    const int wid  = blockIdx.x * (blockDim.x >> 5) + (threadIdx.x >> 5);
    const int tile  = wid / NSPLIT;      // wave-uniform
    const int chunk = wid - tile * NSPLIT;
    const int row0  = tile * 16;
    if (row0 >= nrows) return;           // uniform per wave: branches around all WMMAs

    // Clamp (do not mask): keeps EXEC all-1s for WMMA; garbage rows are never stored.
    int row = row0 + m;
    int rr  = (row < nrows) ? row : (nrows - 1);
    const float* wrow = W + (size_t)rr * (size_t)Kt;

    const int k0    = chunk * KC;                    // multiple of 4
    const int k1    = (k0 + KC < Kt) ? (k0 + KC) : Kt;
    const int kmain = (k1 - k0) & ~3;
    const int kend  = k0 + kmain;
    const int koff  = 2 * half;

    v8f acc = {0.f, 0.f, 0.f, 0.f, 0.f, 0.f, 0.f, 0.f};

    int k = k0;
    // Macro-iteration: 4 independent A/B tile loads issued before 4 chained WMMAs.
    for (; k + 16 <= kend; k += 16) {
        const int kk = k + koff;
        v2f a0 = load_a_tile<EVENK>(wrow, kk);
        v2f a1 = load_a_tile<EVENK>(wrow, kk + 4);
        v2f a2 = load_a_tile<EVENK>(wrow, kk + 8);
        v2f a3 = load_a_tile<EVENK>(wrow, kk + 12);
        v2f b0 = *(const v2f*)(vec + kk);
        v2f b1 = *(const v2f*)(vec + kk + 4);
        v2f b2 = *(const v2f*)(vec + kk + 8);
        v2f b3 = *(const v2f*)(vec + kk + 12);
        acc = wmma_f32_16x16x4(a0, b0, acc);
        acc = wmma_f32_16x16x4(a1, b1, acc);
        acc = wmma_f32_16x16x4(a2, b2, acc);
        acc = wmma_f32_16x16x4(a3, b3, acc);
    }
    for (; k < kend; k += 4) {
        const int kk = k + koff;
        v2f a = load_a_tile<EVENK>(wrow, kk);
        v2f b = *(const v2f*)(vec + kk);
        acc = wmma_f32_16x16x4(a, b, acc);
    }

    if (m == 0) {  // lanes 0 and 16 hold column 0 of D
        #pragma unroll
        for (int i = 0; i < 8; ++i) {
            int r = row0 + half * 8 + i;
            if (r < nrows) {
                float val = acc[i];
                for (int kr = kend; kr < k1; ++kr)   // K remainder (only last chunk of w2)
                    val += W[(size_t)r * (size_t)Kt + kr] * vec[kr];
                part[(size_t)chunk * (size_t)nrows + r] = val;
            }
        }
    }
}

// Reduce NSPLIT partials per row (fixed order -> deterministic), add bias, optional tanh.
__global__ void finalize_kernel(const float* __restrict__ part,
                                const float* __restrict__ bias,
                                float* __restrict__ out,
                                int nrows, int do_tanh) {
    int r = blockIdx.x * blockDim.x + threadIdx.x;
    if (r >= nrows) return;
    float v = bias[r];
    #pragma unroll
    for (int c = 0; c < NSPLIT; ++c)
        v += part[(size_t)c * (size_t)nrows + r];
    out[r] = do_tanh ? tanhf(v) : v;
}

extern "C" void kernel_launch(void* const* d_in, const int* in_sizes, int n_in,
                              void* d_out, int out_size, void* d_ws, size_t ws_size,
                              hipStream_t stream) {
    const float* quotes   = (const float*)d_in[0];
    const float* position = (const float*)d_in[1];
    const float* w_ih_l0  = (const float*)d_in[2];
    // d_in[3] = w_hh_l0 (dead in reference)
    const float* b_ih_l0  = (const float*)d_in[4];
    const float* b_hh_l0  = (const float*)d_in[5];
    const float* w_ih_l1  = (const float*)d_in[6];
    const float* b_ih_l1  = (const float*)d_in[8];
    const float* b_hh_l1  = (const float*)d_in[9];
    const float* w_ih_l2  = (const float*)d_in[10];
    const float* b_ih_l2  = (const float*)d_in[12];
    const float* b_hh_l2  = (const float*)d_in[13];
    const float* w1       = (const float*)d_in[14];
    const float* b1       = (const float*)d_in[15];
    const float* w2       = (const float*)d_in[16];
    const float* b2       = (const float*)d_in[17];

    float* ws    = (float*)d_ws;
    float* conc  = ws;                       // 10000 (16B-aligned base)
    float* vecC  = conc + HID;               // 6004: lin(5000) | position(1001) | pad(3)
    float* part1 = vecC + (KCAT + 3);        // NSPLIT * 5000
    float* part2 = part1 + NSPLIT * MID;     // NSPLIT * 2001
    // total: 10000 + 6004 + 160000 + 64032 = 240036 floats (~940 KB)

    // K-chunk sizes (multiples of 4)
    const int KC1 = (((HID  + NSPLIT - 1) / NSPLIT) + 3) & ~3;  // 316
    const int KC2 = (((KCAT + NSPLIT - 1) / NSPLIT) + 3) & ~3;  // 188

    // Stage 1: LSTM head + combined-vector tail (position + pad)
    lstm_head_kernel<<<4, 256, 0, stream>>>(
        quotes, position,
        w_ih_l0, b_ih_l0, b_hh_l0,
        w_ih_l1, b_ih_l1, b_hh_l1,
        w_ih_l2, b_ih_l2, b_hh_l2,
        conc, vecC + MID);

    // Stage 2: lin = tanh(w1 @ conc + b1). 313 tiles x 32 chunks = 10016 waves.
    {
        int tiles  = (MID + 15) / 16;                    // 313
        int waves  = tiles * NSPLIT;                     // 10016
        int blocks = (waves + 7) / 8;                    // 8 waves / 256-thread block
        gemv_wmma_partial<true><<<blocks, 256, 0, stream>>>(w1, conc, part1, MID, HID, KC1);
        finalize_kernel<<<(MID + 255) / 256, 256, 0, stream>>>(part1, b1, vecC, MID, 1);
    }

    // Stage 3: out = w2 @ [lin; position] + b2. 126 tiles x 32 chunks = 4032 waves.
    {
        int tiles  = (NOUT + 15) / 16;                   // 126
        int waves  = tiles * NSPLIT;                     // 4032
        int blocks = (waves + 7) / 8;
        gemv_wmma_partial<false><<<blocks, 256, 0, stream>>>(w2, vecC, part2, NOUT, KCAT, KC2);
        finalize_kernel<<<(NOUT + 255) / 256, 256, 0, stream>>>(part2, b2, (float*)d_out, NOUT, 0);
    }
}